// opt_loss_54150947668659
// MI455X (gfx1250) — compile-verified
//
#include <hip/hip_runtime.h>
#include <hip/hip_bf16.h>

// ---------------------------------------------------------------------------
// MI455X (gfx1250) implementation.
// Phase 1: single-wave serial forward Riccati scan (register-resident state),
//          stores reversed Gs/Xis/zetas to d_out and packed 48B backward
//          records {Xi, A, zeta, b} to d_ws.
// Phase 2: single-wave serial backward z scan over the records (L2-resident,
//          software prefetch ahead on the serial chain), stores zs.
// Phase 3: data-parallel recompute of pbars/vbars with fp32 WMMA
//          (V_WMMA_F32_16X16X4_F32), 8 independent 2x2 problems block-diagonal
//          embedded per WMMA; exact fp32 math (zero padding contributes 0).
// ---------------------------------------------------------------------------

typedef __attribute__((ext_vector_type(2))) float v2f;
typedef __attribute__((ext_vector_type(8))) float v8f;

struct M2 { float a, b, c, d; };
struct V2 { float x, y; };

static __device__ __forceinline__ M2 mmul(M2 p, M2 q) {
    M2 r;
    r.a = p.a * q.a + p.b * q.c;
    r.b = p.a * q.b + p.b * q.d;
    r.c = p.c * q.a + p.d * q.c;
    r.d = p.c * q.b + p.d * q.d;
    return r;
}
static __device__ __forceinline__ V2 mv(M2 p, V2 v) {
    V2 r; r.x = p.a * v.x + p.b * v.y; r.y = p.c * v.x + p.d * v.y; return r;
}
static __device__ __forceinline__ M2 madd2(M2 p, M2 q) {
    M2 r; r.a = p.a + q.a; r.b = p.b + q.b; r.c = p.c + q.c; r.d = p.d + q.d; return r;
}
static __device__ __forceinline__ M2 inv2(M2 m) {
    float det = m.a * m.d - m.b * m.c;
    M2 r; r.a = m.d / det; r.b = -m.b / det; r.c = -m.c / det; r.d = m.a / det;
    return r;
}

// ---------------------------------------------------------------------------
// Phase 1: forward scan (serial, one wave, lanes redundant, lane 0 stores)
// ---------------------------------------------------------------------------
__global__ void fwd_scan_kernel(const float* __restrict__ Sig, const float* __restrict__ Pi,
                                const float* __restrict__ Kin, const float* __restrict__ Q,
                                const float* __restrict__ P, const float* __restrict__ R,
                                const float* __restrict__ S, const float* __restrict__ gamma,
                                const float* __restrict__ l, const float* __restrict__ rp,
                                const float* __restrict__ dtp, const int* __restrict__ n2p,
                                float* __restrict__ out, float* __restrict__ ws) {
    const int   N2 = n2p[0];
    const float r  = rp[0];
    const float dt = dtp[0];
    const float l0 = l[0], l1 = l[1];

    M2 Sg = {Sig[0], Sig[1], Sig[2], Sig[3]};
    M2 PiM = {Pi[0], Pi[1], Pi[2], Pi[3]};
    M2 KM  = {Kin[0], Kin[1], Kin[2], Kin[3]};
    M2 QM  = {Q[0], Q[1], Q[2], Q[3]};
    M2 PM  = {P[0], P[1], P[2], P[3]};

    M2 Sig2 = mmul(Sg, Sg);
    M2 PimK = {PiM.a - KM.a, PiM.b - KM.b, PiM.c - KM.c, PiM.d - KM.d};
    M2 ImS  = {1.f - S[0], -S[1], -S[2], 1.f - S[3]};
    M2 QS   = mmul(QM, ImS);
    M2 ImR  = {1.f - R[0], -R[1], -R[2], 1.f - R[3]};
    M2 PR   = mmul(PM, ImR);
    M2 KK   = mmul(KM, KM);
    M2 L2   = {PimK.a + l0, PimK.b + l0, PimK.c + l1, PimK.d + l1};  // l + PimK (broadcast)

    M2 G  = QM;         // Gamma_0 = Q
    M2 Xi = QS;         // Xi_0 = Q(I-S)
    V2 ze = {-gamma[0], -gamma[1]};

    float* Gs  = out;
    float* Xis = out + (size_t)N2 * 4;
    float* zes = out + (size_t)N2 * 8;

    for (int t = 0; t < N2; ++t) {
        // A, b from current Gamma
        M2 SG   = mmul(Sig2, G);
        M2 Minv = inv2(madd2(SG, PR));
        M2 A    = mmul(KM, Minv);
        M2 SM   = mmul(Sig2, Minv);
        M2 Bm   = mmul(SM, G);
        V2 bv   = {Bm.a, Bm.d};   // diag(Sig2 @ Minv @ Gamma)

        const int idx = N2 - 1 - t;  // store reversed-in-time directly
        if (threadIdx.x == 0) {
            ((float4*)Gs)[idx]  = make_float4(G.a, G.b, G.c, G.d);
            ((float4*)Xis)[idx] = make_float4(Xi.a, Xi.b, Xi.c, Xi.d);
            ((float2*)zes)[idx] = make_float2(ze.x, ze.y);
            float* rec = ws + (size_t)idx * 12;
            ((float4*)rec)[0] = make_float4(Xi.a, Xi.b, Xi.c, Xi.d);
            ((float4*)rec)[1] = make_float4(A.a, A.b, A.c, A.d);
            ((float4*)rec)[2] = make_float4(ze.x, ze.y, bv.x, bv.y);
        }
        if (t == N2 - 1) break;

        // Gamma update: Gamma + (-((KK@G)@G)@inv(Sig2@G + P) + 2r*G + Q)*dt
        M2 Minv2 = inv2(madd2(SG, PM));
        M2 T = mmul(KK, G);
        T = mmul(T, G);
        T = mmul(T, Minv2);
        M2 Gn;
        Gn.a = G.a + (-T.a + 2.f * r * G.a + QM.a) * dt;
        Gn.b = G.b + (-T.b + 2.f * r * G.b + QM.b) * dt;
        Gn.c = G.c + (-T.c + 2.f * r * G.c + QM.c) * dt;
        Gn.d = G.d + (-T.d + 2.f * r * G.d + QM.d) * dt;

        // zeta update: zeta + ((r*one + Xi@PimK@A)@zeta + Xi@(l+PimK)@b)*dt
        M2 XP = mmul(Xi, PimK);
        M2 M1 = mmul(XP, A);
        V2 t1;
        t1.x = (r + M1.a) * ze.x + (r + M1.b) * ze.y;
        t1.y = (r + M1.c) * ze.x + (r + M1.d) * ze.y;
        M2 XL = mmul(Xi, L2);
        V2 t2 = mv(XL, bv);
        V2 zen = {ze.x + (t1.x + t2.x) * dt, ze.y + (t1.y + t2.y) * dt};

        // Xi update: Xi + ((Xi@PimK@A)@Xi + 2r*Xi + QS)*dt
        M2 M1X = mmul(M1, Xi);
        M2 Xin;
        Xin.a = Xi.a + (M1X.a + 2.f * r * Xi.a + QS.a) * dt;
        Xin.b = Xi.b + (M1X.b + 2.f * r * Xi.b + QS.b) * dt;
        Xin.c = Xi.c + (M1X.c + 2.f * r * Xi.c + QS.c) * dt;
        Xin.d = Xi.d + (M1X.d + 2.f * r * Xi.d + QS.d) * dt;

        G = Gn; Xi = Xin; ze = zen;
    }
}

// ---------------------------------------------------------------------------
// Phase 2: backward z scan (serial, one wave), stores zs
// ---------------------------------------------------------------------------
__global__ void bwd_scan_kernel(const float* __restrict__ Pi, const float* __restrict__ Kin,
                                const float* __restrict__ l, const float* __restrict__ rp,
                                const float* __restrict__ dtp, const float* __restrict__ x0,
                                const int* __restrict__ n2p,
                                float* __restrict__ out, const float* __restrict__ ws) {
    const int   N2 = n2p[0];
    const float r  = rp[0];
    const float dt = dtp[0];
    M2 PimK = {Pi[0] - Kin[0], Pi[1] - Kin[1], Pi[2] - Kin[2], Pi[3] - Kin[3]};
    const float l0 = l[0], l1 = l[1];

    float* zs = out + (size_t)N2 * 10;   // after Gs(4N)+Xis(4N)+zetas(2N)
    V2 z = {x0[0], x0[1]};
    if (threadIdx.x == 0) ((float2*)zs)[0] = make_float2(z.x, z.y);

    for (int k = 0; k < N2; ++k) {
        const float* rec = ws + (size_t)k * 12;
        if (k + 8 < N2) __builtin_prefetch(ws + (size_t)(k + 8) * 12, 0, 0);
        float4 f0 = ((const float4*)rec)[0];   // Xi
        float4 f1 = ((const float4*)rec)[1];   // A
        float4 f2 = ((const float4*)rec)[2];   // zeta, b
        V2 pbar;
        pbar.x = f0.x * z.x + f0.y * z.y + f2.x;
        pbar.y = f0.z * z.x + f0.w * z.y + f2.y;
        V2 vbar;
        vbar.x = f1.x * pbar.x + f1.y * pbar.y + f2.z;
        vbar.y = f1.z * pbar.x + f1.w * pbar.y + f2.w;
        V2 Pv = mv(PimK, vbar);
        z.x = z.x + (r * z.x + l0 + Pv.x) * dt;
        z.y = z.y + (r * z.y + l1 + Pv.y) * dt;
        if (threadIdx.x == 0) ((float2*)zs)[k + 1] = make_float2(z.x, z.y);
    }
}

// ---------------------------------------------------------------------------
// Phase 3: pbars/vbars via fp32 WMMA. 8 independent 2x2 problems per WMMA:
//   A (16x4): Xi_p in rows 2p..2p+1, cols 0..1 (rest 0)
//   B (4x16): z_p in rows 0..1 of column p (rest 0)
//   C (16x16): zeta_p at (2p+i, p)
// => D[2p+i, p] = pbar_p[i], exact fp32 (zero padding adds exactly 0).
// Second WMMA does vbar = A_p @ pbar_p + b_p (pbar handed off via LDS).
// A layout: lane=M (0..15 low half), vgpr=K (K=0/1 low lanes; K=2/3 in high
// lanes, kept zero). B layout mirrored: lane=N, vgpr=K. C/D per ISA 7.12.2.
// ---------------------------------------------------------------------------
__global__ void pv_wmma_kernel(const int* __restrict__ n2p,
                               float* __restrict__ out, const float* __restrict__ ws) {
    const int N2 = n2p[0];
    const float* recs  = ws;
    const float* zs    = out + (size_t)N2 * 10;
    float* pbars = out + (size_t)N2 * 10 + (size_t)(N2 + 1) * 2;
    float* vbars = pbars + (size_t)N2 * 2;

    __shared__ float lds_pbar[16];

    const int  lane = (int)threadIdx.x;
    const int  n    = (lane < 16) ? lane : lane - 16;   // column index this lane holds in C/D
    const bool hi   = lane >= 16;

    const int nch = N2 >> 3;  // 8 problems per chunk
    for (int c0 = (int)blockIdx.x; c0 < nch; c0 += (int)gridDim.x) {
        const int k0 = c0 << 3;

        // ---- WMMA 1: pbar = Xi @ z + zeta ----
        v2f va; va.x = 0.f; va.y = 0.f;
        if (lane < 16) {
            const int p = lane >> 1, i = lane & 1;
            const float* rec = recs + (size_t)(k0 + p) * 12;
            va.x = rec[i * 2 + 0];   // Xi[i][0] -> A[2p+i, 0]
            va.y = rec[i * 2 + 1];   // Xi[i][1] -> A[2p+i, 1]
        }
        v2f vb; vb.x = 0.f; vb.y = 0.f;
        if (lane < 8) {
            vb.x = zs[(size_t)(k0 + lane) * 2 + 0];  // B[0, p] = z_p[0]
            vb.y = zs[(size_t)(k0 + lane) * 2 + 1];  // B[1, p] = z_p[1]
        }
        v8f vc = {0.f, 0.f, 0.f, 0.f, 0.f, 0.f, 0.f, 0.f};
        if (n < 8) {
            const float* rec = recs + (size_t)(k0 + n) * 12;
#pragma unroll
            for (int v = 0; v < 8; ++v) {
                const int m = hi ? v + 8 : v;
                if ((m >> 1) == n) vc[v] = rec[8 + (m & 1)];  // zeta_n[i]
            }
        }
        v8f d = __builtin_amdgcn_wmma_f32_16x16x4_f32(false, va, false, vb,
                                                      (short)0, vc, false, false);
        // extract pbar_p (owners: lanes 0..3 and 20..23)
#pragma unroll
        for (int v = 0; v < 8; ++v) {
            const int m = hi ? v + 8 : v;
            if (n < 8 && (m >> 1) == n) {
                const float val = d[v];
                pbars[(size_t)(k0 + n) * 2 + (m & 1)] = val;
                lds_pbar[n * 2 + (m & 1)] = val;
            }
        }
        __syncthreads();

        // ---- WMMA 2: vbar = A @ pbar + b ----
        v2f va2; va2.x = 0.f; va2.y = 0.f;
        if (lane < 16) {
            const int p = lane >> 1, i = lane & 1;
            const float* rec = recs + (size_t)(k0 + p) * 12 + 4;  // A matrix
            va2.x = rec[i * 2 + 0];
            va2.y = rec[i * 2 + 1];
        }
        v2f vb2; vb2.x = 0.f; vb2.y = 0.f;
        if (lane < 8) {
            vb2.x = lds_pbar[lane * 2 + 0];
            vb2.y = lds_pbar[lane * 2 + 1];
        }
        v8f vc2 = {0.f, 0.f, 0.f, 0.f, 0.f, 0.f, 0.f, 0.f};
        if (n < 8) {
            const float* rec = recs + (size_t)(k0 + n) * 12;
#pragma unroll
            for (int v = 0; v < 8; ++v) {
                const int m = hi ? v + 8 : v;
                if ((m >> 1) == n) vc2[v] = rec[10 + (m & 1)];  // b_n[i]
            }
        }
        v8f d2 = __builtin_amdgcn_wmma_f32_16x16x4_f32(false, va2, false, vb2,
                                                       (short)0, vc2, false, false);
#pragma unroll
        for (int v = 0; v < 8; ++v) {
            const int m = hi ? v + 8 : v;
            if (n < 8 && (m >> 1) == n) {
                vbars[(size_t)(k0 + n) * 2 + (m & 1)] = d2[v];
            }
        }
        __syncthreads();  // protect lds_pbar before next chunk
    }
}

// ---------------------------------------------------------------------------
extern "C" void kernel_launch(void* const* d_in, const int* in_sizes, int n_in,
                              void* d_out, int out_size, void* d_ws, size_t ws_size,
                              hipStream_t stream) {
    (void)in_sizes; (void)n_in; (void)out_size; (void)ws_size;
    const float* Sig   = (const float*)d_in[0];
    const float* Pi    = (const float*)d_in[1];
    const float* K     = (const float*)d_in[2];
    const float* mat_Q = (const float*)d_in[3];
    const float* mat_P = (const float*)d_in[4];
    const float* mat_R = (const float*)d_in[5];
    const float* mat_S = (const float*)d_in[6];
    const float* gamma = (const float*)d_in[7];
    const float* l     = (const float*)d_in[8];
    const float* x0    = (const float*)d_in[9];
    const float* r     = (const float*)d_in[10];
    const float* del_t = (const float*)d_in[11];
    const int*   n2p   = (const int*)d_in[12];

    float* out = (float*)d_out;
    float* ws  = (float*)d_ws;

    // Serial scans: one wave each (latency-bound dependency chains).
    fwd_scan_kernel<<<1, 32, 0, stream>>>(Sig, Pi, K, mat_Q, mat_P, mat_R, mat_S,
                                          gamma, l, r, del_t, n2p, out, ws);
    bwd_scan_kernel<<<1, 32, 0, stream>>>(Pi, K, l, r, del_t, x0, n2p, out, ws);
    // Parallel WMMA pass over all 16384 steps (grid-stride over N2/8 chunks).
    pv_wmma_kernel<<<256, 32, 0, stream>>>(n2p, out, ws);
}